// MambaBlockTVM_46643344834628
// MI455X (gfx1250) — compile-verified
//
#include <hip/hip_runtime.h>
#include <hip/hip_bf16.h>

// ---------------------------------------------------------------------------
// Mamba block for MI455X (gfx1250, CDNA5, wave32).
// GEMMs: V_WMMA_F32_16X16X4_F32, 64x64 wave tiles, chunked register staging.
// Scan:  register-resident recurrence (16 states/thread) with the shared
//        B/C row staged into LDS via GLOBAL_LOAD_ASYNC_TO_LDS (double
//        buffered, ASYNCcnt-tracked) -- removes 256x-redundant global reads.
// Workspace layout (floats), total ~84.7 MB:
//   xz[2048,4096] xc[2048,2048] x_dbl[2048,96] delta[2048,2048] y2[2048,2048]
// ---------------------------------------------------------------------------

#define B_SZ   2
#define L_SZ   1024
#define DM     1024
#define DI     2048
#define NSTATE 16
#define DTR    64
#define XDBL   96   // DTR + 2*NSTATE

typedef float v2f __attribute__((ext_vector_type(2)));
typedef float v8f __attribute__((ext_vector_type(8)));

// ---------------------------------------------------------------------------
// Generic fp32 WMMA GEMM:   C[M,N] = A[M,K] * W[N,K]^T   (+ bias, + softplus)
// One wave computes a (16*MT) x (16*NT) output tile; K stepped by 4 per
// v_wmma_f32_16x16x4_f32, staged KU k-steps at a time so all global_load_b64
// for a chunk issue before the WMMA batch (amortized s_wait_loadcnt).
// Fragment layouts (CDNA5 ISA 7.12.2, 32-bit):
//   A 16x4 : lanes 0-15 -> rows M=0..15 ; VGPR0 = K[2*half], VGPR1 = K[2*half+1]
//   B 4x16 : lanes 0-15 -> cols N=0..15 ; VGPR0 = K[2*half], VGPR1 = K[2*half+1]
//   C 16x16: lane&15 -> N ; VGPR i -> M = i + 8*(lane>>4)
// EPI: 0 = none, 1 = +bias then softplus.  Requires K % (4*KU) == 0.
// ---------------------------------------------------------------------------
template <int MT, int NT, int EPI>
__global__ __launch_bounds__(128) void mamba_wmma_gemm_f32(
    const float* __restrict__ A, int lda,
    const float* __restrict__ W, int ldw,
    const float* __restrict__ bias,
    float* __restrict__ C, int ldc,
    int M, int N, int K)
{
  constexpr int KU = 2;                       // k-steps staged per chunk
  const int wid    = blockIdx.x * (blockDim.x >> 5) + (threadIdx.x >> 5);
  const int mgrp   = M / (16 * MT);
  const int ngrp   = N / (16 * NT);
  if (wid >= mgrp * ngrp) return;             // wave-uniform: EXEC stays all-1s

  const int mg   = wid % mgrp;
  const int ng   = wid / mgrp;
  const int lane = threadIdx.x & 31;
  const int rl   = lane & 15;                 // row (A) / col (B,C) in subtile
  const int half = lane >> 4;
  const int m0   = mg * (16 * MT);
  const int n0   = ng * (16 * NT);

  const float* Arow[MT];
#pragma unroll
  for (int mt = 0; mt < MT; ++mt)
    Arow[mt] = A + (size_t)(m0 + 16 * mt + rl) * lda + 2 * half;
  const float* Wrow[NT];
#pragma unroll
  for (int t = 0; t < NT; ++t)
    Wrow[t] = W + (size_t)(n0 + 16 * t + rl) * ldw + 2 * half;

  v8f acc[MT][NT] = {};

  for (int k0 = 0; k0 < K; k0 += 4 * KU) {
    // Stage the whole chunk into registers first: loads batch, partial
    // s_wait_loadcnt covers the WMMA batch.
    v2f a[KU][MT];
    v2f b[KU][NT];
#pragma unroll
    for (int u = 0; u < KU; ++u)
#pragma unroll
      for (int mt = 0; mt < MT; ++mt)
        a[u][mt] = *(const v2f*)(Arow[mt] + k0 + 4 * u);
#pragma unroll
    for (int u = 0; u < KU; ++u)
#pragma unroll
      for (int t = 0; t < NT; ++t)
        b[u][t] = *(const v2f*)(Wrow[t] + k0 + 4 * u);

    // Near-cache speculative prefetch (global_prefetch_b8), 16 k-steps ahead.
    __builtin_prefetch(Arow[0] + k0 + 64, 0, 3);
    __builtin_prefetch(Wrow[0] + k0 + 64, 0, 3);

#pragma unroll
    for (int u = 0; u < KU; ++u)
#pragma unroll
      for (int mt = 0; mt < MT; ++mt)
#pragma unroll
        for (int t = 0; t < NT; ++t)
          acc[mt][t] = __builtin_amdgcn_wmma_f32_16x16x4_f32(
              /*neg_a=*/false, a[u][mt], /*neg_b=*/false, b[u][t],
              /*c_mod=*/(short)0, acc[mt][t],
              /*reuse_a=*/false, /*reuse_b=*/false);
  }

#pragma unroll
  for (int mt = 0; mt < MT; ++mt) {
#pragma unroll
    for (int t = 0; t < NT; ++t) {
      const int col = n0 + 16 * t + rl;
      const float bv = (EPI == 1) ? bias[col] : 0.0f;
#pragma unroll
      for (int i = 0; i < 8; ++i) {
        const int row = m0 + 16 * mt + 8 * half + i;
        float v = acc[mt][t][i];
        if (EPI == 1) {
          v += bv;
          v = (v > 20.0f) ? v : __logf(1.0f + __expf(v));  // softplus
        }
        C[(size_t)row * ldc + col] = v;
      }
    }
  }
}

// ---------------------------------------------------------------------------
// Depthwise causal conv1d (k=4, left pad 3) + bias + SiLU.
// One thread per (b,l,d). xi = xz[..., 0:DI].
// ---------------------------------------------------------------------------
__global__ __launch_bounds__(256) void mamba_conv_silu(
    const float* __restrict__ xz,
    const float* __restrict__ conv_w,   // [DI,1,4]
    const float* __restrict__ conv_b,   // [DI]
    float* __restrict__ xc)             // [B,L,DI]
{
  const int idx = blockIdx.x * blockDim.x + threadIdx.x;
  if (idx >= B_SZ * L_SZ * DI) return;
  const int d = idx & (DI - 1);
  const int l = (idx >> 11) & (L_SZ - 1);
  const int b = idx >> 21;

  const float* w = conv_w + d * 4;
  float acc = conv_b[d];
#pragma unroll
  for (int j = 0; j < 4; ++j) {
    const int li = l - 3 + j;
    if (li >= 0)
      acc += w[j] * xz[(size_t)(b * L_SZ + li) * (2 * DI) + d];
  }
  xc[idx] = acc / (1.0f + __expf(-acc));   // SiLU
}

// ---------------------------------------------------------------------------
// CDNA5 async global->LDS copy (ASYNCcnt-tracked).  One b32 per enabled lane:
//   dword at the lane's vaddr -> LDS at the lane's vdst offset        (VGLOBAL GLOBAL_LOAD_ASYNC_TO_LDS)
// LDS address = low 32 bits of the generic pointer (flat->LDS truncation,
// ISA section 10.2).
// ---------------------------------------------------------------------------
static __device__ __forceinline__ void async_ld_lds_b32(uint32_t lds_off,
                                                        const float* gptr) {
  asm volatile("global_load_async_to_lds_b32 %0, %1, off"
               :: "v"(lds_off), "v"((uint64_t)(uintptr_t)gptr)
               : "memory");
}

static __device__ __forceinline__ void wait_asynccnt_le1() {
#if __has_builtin(__builtin_amdgcn_s_wait_asynccnt)
  __builtin_amdgcn_s_wait_asynccnt(1);
#else
  asm volatile("s_wait_asynccnt 0x1" ::: "memory");
#endif
}

// ---------------------------------------------------------------------------
// Selective scan. One thread per (b,d): h[16] in registers, sequential in L.
// All 256 threads of a block share (b,l), so the 32-float B/C row of x_dbl is
// staged once per timestep into LDS by wave 0 via async load (double
// buffered: load for l+1 overlaps compute for l), instead of 256 redundant
// global reads. Fused epilogue: y2 = (y + xc*D) * silu(z).
// ---------------------------------------------------------------------------
__global__ __launch_bounds__(256) void mamba_scan(
    const float* __restrict__ delta,   // [B,L,DI]
    const float* __restrict__ xc,      // [B,L,DI]
    const float* __restrict__ x_dbl,   // [B,L,96] (B at +64, C at +80)
    const float* __restrict__ xz,      // [B,L,2*DI] (z at +DI)
    const float* __restrict__ A_log,   // [DI,16]
    const float* __restrict__ Dp,      // [DI]
    float* __restrict__ y2)            // [B,L,DI]
{
  __shared__ float sbc[2][32];         // [buffer][B(0:16) | C(16:32)]

  const int tid = threadIdx.x;
  const int t   = blockIdx.x * blockDim.x + tid;
  const int b   = t / DI;              // uniform per block (8 blocks per b)
  const int d   = t % DI;

  float Ac[NSTATE];
#pragma unroll
  for (int n = 0; n < NSTATE; ++n)
    Ac[n] = -__expf(A_log[d * NSTATE + n]);
  const float Dv = Dp[d];

  const float* xdbl_b = x_dbl + (size_t)b * L_SZ * XDBL + DTR;  // B/C rows
  const uint32_t off0 = (uint32_t)(uintptr_t)&sbc[0][tid & 31];
  const uint32_t off1 = (uint32_t)(uintptr_t)&sbc[1][tid & 31];

  // Prologue: wave 0 stages B/C for l = 0 into buffer 0.
  if (tid < 32)
    async_ld_lds_b32(off0, xdbl_b + tid);

  float h[NSTATE] = {};

  for (int l = 0; l < L_SZ; ++l) {
    const int buf = l & 1;
    __syncthreads();                   // all waves done with buffer buf^1
    if (tid < 32) {
      // Issue load for l+1 into the other buffer (duplicate on last step so
      // the <=1 wait below always covers the current buffer's load).
      const int ln = (l + 1 < L_SZ) ? l + 1 : L_SZ - 1;
      async_ld_lds_b32(buf ? off0 : off1, xdbl_b + (size_t)ln * XDBL + tid);
      wait_asynccnt_le1();             // in-order: load for step l complete
    }
    __syncthreads();                   // publish buffer buf

    const size_t bl  = (size_t)b * L_SZ + l;
    const float  dlt = delta[bl * DI + d];
    const float  xcv = xc[bl * DI + d];
    const float  zv  = xz[bl * (2 * DI) + DI + d];
    const float* bc  = &sbc[buf][0];   // broadcast reads, conflict-free

    float y = 0.0f;
    const float dx = dlt * xcv;
#pragma unroll
    for (int n = 0; n < NSTATE; ++n) {
      const float aa = __expf(dlt * Ac[n]);
      h[n] = aa * h[n] + dx * bc[n];
      y += h[n] * bc[NSTATE + n];
    }
    y += xcv * Dv;
    const float sz = zv / (1.0f + __expf(-zv));  // silu(z)
    y2[bl * DI + d] = y * sz;
  }
}

// ---------------------------------------------------------------------------
extern "C" void kernel_launch(void* const* d_in, const int* in_sizes, int n_in,
                              void* d_out, int out_size, void* d_ws, size_t ws_size,
                              hipStream_t stream) {
  (void)in_sizes; (void)n_in; (void)out_size; (void)ws_size;

  const float* x         = (const float*)d_in[0];  // [B,L,DM]
  const float* in_proj_w = (const float*)d_in[1];  // [2*DI, DM]
  const float* conv_w    = (const float*)d_in[2];  // [DI,1,4]
  const float* conv_b    = (const float*)d_in[3];  // [DI]
  const float* x_proj_w  = (const float*)d_in[4];  // [96, DI]
  const float* dt_proj_w = (const float*)d_in[5];  // [DI, 64]
  const float* dt_proj_b = (const float*)d_in[6];  // [DI]
  const float* A_log     = (const float*)d_in[7];  // [DI, 16]
  const float* Dvec      = (const float*)d_in[8];  // [DI]
  const float* out_proj_w= (const float*)d_in[9];  // [DM, DI]
  float*       out       = (float*)d_out;          // [B,L,DM]

  const int M = B_SZ * L_SZ;                       // 2048 rows everywhere

  float* ws    = (float*)d_ws;
  float* xz    = ws;                               // M * 4096
  float* xc    = xz    + (size_t)M * (2 * DI);     // M * 2048
  float* xdbl  = xc    + (size_t)M * DI;           // M * 96
  float* delta = xdbl  + (size_t)M * XDBL;         // M * 2048
  float* y2    = delta + (size_t)M * DI;           // M * 2048

  // 1) in_proj: xz[M,4096] = x[M,1024] * in_proj_w[4096,1024]^T
  {
    const int waves = (M / 64) * ((2 * DI) / 64);  // MT=4, NT=4 -> 2048 waves
    mamba_wmma_gemm_f32<4, 4, 0><<<(waves + 3) / 4, 128, 0, stream>>>(
        x, DM, in_proj_w, DM, nullptr, xz, 2 * DI, M, 2 * DI, DM);
  }

  // 2) depthwise causal conv + SiLU -> xc[M,2048]
  mamba_conv_silu<<<(B_SZ * L_SZ * DI) / 256, 256, 0, stream>>>(
      xz, conv_w, conv_b, xc);

  // 3) x_proj: xdbl[M,96] = xc[M,2048] * x_proj_w[96,2048]^T
  {
    const int waves = (M / 64) * (XDBL / 32);      // MT=4, NT=2 -> 96 waves
    mamba_wmma_gemm_f32<4, 2, 0><<<(waves + 3) / 4, 128, 0, stream>>>(
        xc, DI, x_proj_w, DI, nullptr, xdbl, XDBL, M, XDBL, DI);
  }

  // 4) dt_proj + bias + softplus: delta[M,2048] = sp(xdbl[:, :64] * Wdt^T + b)
  {
    const int waves = (M / 64) * (DI / 64);        // MT=4, NT=4 -> 1024 waves
    mamba_wmma_gemm_f32<4, 4, 1><<<(waves + 3) / 4, 128, 0, stream>>>(
        xdbl, XDBL, dt_proj_w, DTR, dt_proj_b, delta, DI, M, DI, DTR);
  }

  // 5) selective scan (+ xc*D, * silu(z)) -> y2[M,2048]
  mamba_scan<<<(B_SZ * DI) / 256, 256, 0, stream>>>(
      delta, xc, xdbl, xz, A_log, Dvec, y2);

  // 6) out_proj: out[M,1024] = y2[M,2048] * out_proj_w[1024,2048]^T
  {
    const int waves = (M / 64) * (DM / 64);        // MT=4, NT=4 -> 512 waves
    mamba_wmma_gemm_f32<4, 4, 0><<<(waves + 3) / 4, 128, 0, stream>>>(
        y2, DI, out_proj_w, DI, nullptr, out, DM, M, DM, DI);
  }
}